// SSIMLoss_48979807043793
// MI455X (gfx1250) — compile-verified
//
#include <hip/hip_runtime.h>

// ---------------------------------------------------------------------------
// SSIM loss, fully matrix-pipe conv: BOTH separable Gaussian passes run as
// banded-weight matmuls on V_WMMA_F32_16X16X4_F32. Per 16x16 tile:
//   horizontal: H(16x16) = In(16x28) x W2(28x16),  W2[k][n] = g[k-n]
//   vertical:   D(16x16) = W1(16x28) x H(28x16),   W1[m][k] = g[k-m]
// Under the mirrored A/B fragment layouts both band operands reduce to the
// SAME per-lane values g[k - (lane&15)], loaded once from a precomputed
// table in d_ws. f32 throughout (variance terms are cancellation-sensitive).
// ---------------------------------------------------------------------------

typedef __attribute__((ext_vector_type(2))) float v2f;
typedef __attribute__((ext_vector_type(8))) float v8f;

#define IMG    512
#define TILE   16
#define HALO   5
#define REG    26     // TILE + 2*HALO valid rows/cols per tile
#define KPAD   28     // K padded to 7 WMMA steps of 4
#define RROWS  32     // raw rows padded so group-1 A reads stay in-bounds
#define NSTEP  7
#define WAVES  4      // waves (tiles) per 128-thread block
#define NTOT   25165824.0   // 32*3*512*512
#define WT_OFF 16     // byte offset of band table inside d_ws

// Normalized Gaussian, sigma = 1.5, 11 taps (symmetric).
__device__ constexpr float GW[11] = {
    0.00102840f, 0.00759877f, 0.03600077f, 0.10936100f, 0.21300552f,
    0.26601245f,
    0.21300552f, 0.10936100f, 0.03600077f, 0.00759877f, 0.00102840f};

// Banded weight g[u] for 0 <= u <= 10 else 0; select chain -> no scratch.
__device__ __forceinline__ float band_w(int u) {
    float w = 0.0f;
#pragma unroll
    for (int j = 0; j < 11; ++j) w = (u == j) ? GW[j] : w;
    return w;
}

__device__ __forceinline__ v8f wmma4(v2f a, v2f b, v8f c) {
    return __builtin_amdgcn_wmma_f32_16x16x4_f32(
        false, a, false, b, (short)0, c, false, false);
}

__global__ __launch_bounds__(128) void ssim_fused_kernel(
    const float* __restrict__ pred, const float* __restrict__ targ,
    void* __restrict__ ws) {
    // Per-wave LDS: raw p,t halo tiles + horizontal conv results (5 fields).
    __shared__ float raw[WAVES][2][RROWS][KPAD];   // rows 26..31 unused pad
    __shared__ float Hf[WAVES][5][KPAD][TILE];     // rows 26,27 zeroed
    __shared__ float partial[WAVES];

    double* accum   = (double*)ws;
    const float* Wt = (const float*)((const char*)ws + WT_OFF); // [16][28]

    const int lane = threadIdx.x & 31;
    const int wave = threadIdx.x >> 5;
    const int tile = blockIdx.x * WAVES + wave;   // < 98304 by launch config
    const int tx = tile & 31;
    const int ty = (tile >> 5) & 31;
    const int nc = tile >> 10;                    // fused N*C plane, 0..95
    const int x0 = tx * TILE;
    const int y0 = ty * TILE;
    const size_t plane = (size_t)nc * (IMG * IMG);

    // ---------------- Phase 0: stage raw halo tiles into LDS --------------
    // Lane r (< 26) loads row y0-5+r, cols x0-5 .. x0+22 (28 cols; cols
    // 26,27 are K-pad: band weight is 0 there, any finite value is fine,
    // out-of-image -> 0 (lax.conv zero padding)).
    if (lane < REG) {
        const int y = y0 - HALO + lane;
        const bool vr = (y >= 0) && (y < IMG);
        const float* prow = pred + plane + (size_t)(vr ? y : 0) * IMG;
        const float* trow = targ + plane + (size_t)(vr ? y : 0) * IMG;
#pragma unroll
        for (int c = 0; c < KPAD; ++c) {
            const int x  = x0 - HALO + c;
            const bool v = vr && (x >= 0) && (x < IMG);
            raw[wave][0][lane][c] = v ? prow[x] : 0.0f;
            raw[wave][1][lane][c] = v ? trow[x] : 0.0f;
        }
    }
    // Zero Hf K-pad rows 26,27 (avoid 0*NaN from uninitialized LDS).
    {
        const int pr = REG + (lane >> 4);      // 26 or 27
        const int pc = lane & 15;
#pragma unroll
        for (int f = 0; f < 5; ++f) Hf[wave][f][pr][pc] = 0.0f;
    }

    // Band fragments: value g[k - (lane&15)] at k = 4s + half-offset.
    // Serve as B (horizontal: W2[k][n]) AND A (vertical: W1[m][k]) under the
    // mirrored fragment layouts (A 16x4: lanes 0-15 K={0,1}, lanes 16-31
    // K={2,3} of row M=lane&15; B 4x16 mirrored with N=lane&15).
    const int m    = lane & 15;
    const int koff = (lane < 16) ? 0 : 2;
    v2f band[NSTEP];
#pragma unroll
    for (int s = 0; s < NSTEP; ++s)
        band[s] = *(const v2f*)(Wt + m * KPAD + 4 * s + koff);  // 8B aligned

    // ---------------- Phase 1: horizontal conv via WMMA -------------------
    // Two row-groups (rows 0-15, 16-31; rows 26-31 are discarded pad: A-row
    // garbage only pollutes the matching D rows, which we never store).
#pragma unroll
    for (int g = 0; g < 2; ++g) {
        v8f a0 = {}, a1 = {}, a2 = {}, a3 = {}, a4 = {};
#pragma unroll
        for (int s = 0; s < NSTEP; ++s) {
            const int k = 4 * s + koff;
            v2f ap, at;
            ap.x = raw[wave][0][g * 16 + m][k];
            ap.y = raw[wave][0][g * 16 + m][k + 1];
            at.x = raw[wave][1][g * 16 + m][k];
            at.y = raw[wave][1][g * 16 + m][k + 1];
            const v2f ap2 = ap * ap;     // field A-fragments derived in-reg
            const v2f at2 = at * at;
            const v2f apt = ap * at;
            a0 = wmma4(ap,  band[s], a0);   // conv(p)
            a1 = wmma4(at,  band[s], a1);   // conv(t)
            a2 = wmma4(ap2, band[s], a2);   // conv(p*p)
            a3 = wmma4(at2, band[s], a3);   // conv(t*t)
            a4 = wmma4(apt, band[s], a4);   // conv(p*t)
        }
        // D layout: VGPR r -> row g*16 + r + (lane<16 ? 0 : 8), col lane&15.
#pragma unroll
        for (int r = 0; r < 8; ++r) {
            const int row = g * 16 + r + ((lane < 16) ? 0 : 8);
            if (row < REG) {
                Hf[wave][0][row][m] = a0[r];
                Hf[wave][1][row][m] = a1[r];
                Hf[wave][2][row][m] = a2[r];
                Hf[wave][3][row][m] = a3[r];
                Hf[wave][4][row][m] = a4[r];
            }
        }
    }
    // Wave-synchronous: per-wave LDS slab, DS ops in-order within the wave;
    // EXEC reconverges to all-ones before every WMMA.

    // ---------------- Phase 2: vertical conv via WMMA ---------------------
    v8f vacc[5];
#pragma unroll
    for (int f = 0; f < 5; ++f) {
        v8f c = {};
#pragma unroll
        for (int s = 0; s < NSTEP; ++s) {
            const int k = 4 * s + koff;
            v2f b;
            b.x = Hf[wave][f][k][m];
            b.y = Hf[wave][f][k + 1][m];
            c = wmma4(band[s], b, c);
        }
        vacc[f] = c;
    }

    // ---------------- Phase 3: SSIM map + reduction -----------------------
    const float C1 = 1e-4f;   // 0.01^2
    const float C2 = 9e-4f;   // 0.03^2
    float sum = 0.0f;
#pragma unroll
    for (int e = 0; e < 8; ++e) {
        const float mu1  = vacc[0][e];
        const float mu2  = vacc[1][e];
        const float m1s  = mu1 * mu1;
        const float m2s  = mu2 * mu2;
        const float m12  = mu1 * mu2;
        const float sg1  = vacc[2][e] - m1s;
        const float sg2  = vacc[3][e] - m2s;
        const float sg12 = vacc[4][e] - m12;
        const float num  = (2.0f * m12 + C1) * (2.0f * sg12 + C2);
        const float den  = (m1s + m2s + C1) * (sg1 + sg2 + C2) + 1e-8f;
        sum += num / den;
    }

#pragma unroll
    for (int off = 16; off > 0; off >>= 1) sum += __shfl_xor(sum, off, 32);
    if (lane == 0) partial[wave] = sum;
    __syncthreads();
    if (threadIdx.x == 0) {
        double s = (double)partial[0] + (double)partial[1] +
                   (double)partial[2] + (double)partial[3];
        atomicAdd(accum, s);
    }
}

// Zero the accumulator and build the 16x28 banded weight table in d_ws.
__global__ void ssim_init_kernel(void* ws) {
    double* acc = (double*)ws;
    float* Wt   = (float*)((char*)ws + WT_OFF);
    const int i = threadIdx.x;
    if (i == 0) acc[0] = 0.0;
    if (i < 16 * KPAD) {
        const int mm = i / KPAD;
        const int kk = i % KPAD;
        Wt[i] = band_w(kk - mm);
    }
}

__global__ void ssim_finalize_kernel(const void* ws, float* out) {
    out[0] = 1.0f - (float)(((const double*)ws)[0] / NTOT);
}

extern "C" void kernel_launch(void* const* d_in, const int* in_sizes, int n_in,
                              void* d_out, int out_size, void* d_ws,
                              size_t ws_size, hipStream_t stream) {
    const float* pred = (const float*)d_in[0];
    const float* targ = (const float*)d_in[1];
    float* out = (float*)d_out;

    // 32x32 tiles per 512x512 plane, 96 planes -> 98304 wave-tiles,
    // 4 wave-tiles per 128-thread block.
    const int nblocks = (32 * 32 * 96) / WAVES;   // 24576

    ssim_init_kernel<<<1, 448, 0, stream>>>(d_ws);
    ssim_fused_kernel<<<nblocks, 128, 0, stream>>>(pred, targ, d_ws);
    ssim_finalize_kernel<<<1, 1, 0, stream>>>(d_ws, out);
}